// GaussianRasterizer_14929306321112
// MI455X (gfx1250) — compile-verified
//
#include <hip/hip_runtime.h>
#include <hip/hip_bf16.h>
#include <math.h>

#define IMG_H 256
#define IMG_W 256
#define NG    512

typedef __attribute__((ext_vector_type(2))) float v2f;
typedef __attribute__((ext_vector_type(8))) float v8f;
typedef int v4i_vs __attribute__((vector_size(16)));   // matches builtin param type

// Workspace layout: 13 depth-sorted SoA arrays of NG floats each.
//  0 mx, 1 my, 2 inv00, 3 inv01, 4 inv11, 5 opac,
//  6 c0, 7 c1, 8 c2, 9 xmin, 10 xmax, 11 ymin, 12 ymax
#define NARR 13

#if __has_builtin(__builtin_amdgcn_wmma_f32_16x16x4_f32)
#define HAVE_WMMA_F32X4 1
#else
#define HAVE_WMMA_F32X4 0
#endif

#if __has_builtin(__builtin_amdgcn_global_load_async_to_lds_b128)
#define HAVE_ASYNC_LDS 1
#else
#define HAVE_ASYNC_LDS 0
#endif

// ---------------------------------------------------------------------------
// Kernel 1: per-Gaussian projection + EWA covariance + bitonic depth argsort.
// One block of 512 threads; everything lives in LDS until the sorted gather.
// ---------------------------------------------------------------------------
__global__ __launch_bounds__(NG) void preprocess_sort_kernel(
    const float* __restrict__ pos,    // (N,3)
    const float* __restrict__ cov3d,  // (N,3,3)
    const float* __restrict__ opac,   // (N,1)
    const float* __restrict__ col,    // (N,3)
    const float* __restrict__ Km,     // (3,3)
    const float* __restrict__ Rm,     // (3,3)
    const float* __restrict__ tv,     // (3,)
    float* __restrict__ ws)
{
  __shared__ float sp[NARR][NG];
  __shared__ float sz[NG];
  __shared__ int   si[NG];

  const int i = threadIdx.x;

  // --- camera (uniform scalar loads) ---
  float R[9], Kc[9], t3[3];
#pragma unroll
  for (int q = 0; q < 9; ++q) { R[q] = Rm[q]; Kc[q] = Km[q]; }
#pragma unroll
  for (int q = 0; q < 3; ++q) t3[q] = tv[q];
  const float fx = Kc[0], fy = Kc[4];

  // --- project mean ---
  const float p0 = pos[i*3+0], p1 = pos[i*3+1], p2 = pos[i*3+2];
  float mc[3];
#pragma unroll
  for (int r = 0; r < 3; ++r)
    mc[r] = R[r*3+0]*p0 + R[r*3+1]*p1 + R[r*3+2]*p2 + t3[r];
  const float z = mc[2];
  float pr[3];
#pragma unroll
  for (int r = 0; r < 3; ++r)
    pr[r] = Kc[r*3+0]*mc[0] + Kc[r*3+1]*mc[1] + Kc[r*3+2]*mc[2];
  const float invw = 1.0f / pr[2];
  const float mx = pr[0] * invw, my = pr[1] * invw;

  // --- Jacobian of perspective projection ---
  const float iz = 1.0f / z, iz2 = iz * iz;
  float J[2][3] = { { fx*iz, 0.0f, -fx*mc[0]*iz2 },
                    { 0.0f,  fy*iz, -fy*mc[1]*iz2 } };

  // --- cov_cam = R * cov3d * R^T ---
  float RS[3][3], CC[3][3];
#pragma unroll
  for (int r = 0; r < 3; ++r)
#pragma unroll
    for (int c = 0; c < 3; ++c) {
      float s = 0.0f;
#pragma unroll
      for (int k = 0; k < 3; ++k) s += R[r*3+k] * cov3d[i*9 + k*3 + c];
      RS[r][c] = s;
    }
#pragma unroll
  for (int r = 0; r < 3; ++r)
#pragma unroll
    for (int c = 0; c < 3; ++c) {
      float s = 0.0f;
#pragma unroll
      for (int k = 0; k < 3; ++k) s += RS[r][k] * R[c*3+k];
      CC[r][c] = s;
    }

  // --- cov2d = J * cov_cam * J^T + 0.3 I ---
  float JC[2][3];
#pragma unroll
  for (int r = 0; r < 2; ++r)
#pragma unroll
    for (int c = 0; c < 3; ++c)
      JC[r][c] = J[r][0]*CC[0][c] + J[r][1]*CC[1][c] + J[r][2]*CC[2][c];
  const float a = JC[0][0]*J[0][0] + JC[0][1]*J[0][1] + JC[0][2]*J[0][2] + 0.3f;
  const float b = JC[0][0]*J[1][0] + JC[0][1]*J[1][1] + JC[0][2]*J[1][2];
  const float c = JC[1][0]*J[1][0] + JC[1][1]*J[1][1] + JC[1][2]*J[1][2] + 0.3f;

  const float det  = fmaxf(a*c - b*b, 1e-8f);
  const float idet = 1.0f / det;
  const float i00 = c * idet, i01 = -b * idet, i11 = a * idet;

  const float mid = 0.5f * (a - c);
  const float max_eig = 0.5f*(a + c) + sqrtf(fmaxf(mid*mid + b*b, 1e-8f));
  const float radii = 3.0f * sqrtf(max_eig);

  sp[0][i] = mx;  sp[1][i] = my;
  sp[2][i] = i00; sp[3][i] = i01; sp[4][i] = i11;
  sp[5][i] = opac[i];
  sp[6][i] = col[i*3+0]; sp[7][i] = col[i*3+1]; sp[8][i] = col[i*3+2];
  sp[9][i]  = fmaxf(0.0f, truncf(mx - radii));
  sp[10][i] = fminf((float)IMG_W, truncf(mx + radii) + 1.0f);
  sp[11][i] = fmaxf(0.0f, truncf(my - radii));
  sp[12][i] = fminf((float)IMG_H, truncf(my + radii) + 1.0f);
  sz[i] = z;
  si[i] = i;

  // --- bitonic argsort ascending by (z, idx) — matches stable jnp.argsort ---
  for (unsigned k = 2; k <= NG; k <<= 1) {
    for (unsigned j = k >> 1; j > 0; j >>= 1) {
      __syncthreads();
      unsigned tid = (unsigned)i;
      unsigned ixj = tid ^ j;
      if (ixj > tid) {
        float z1 = sz[tid], z2 = sz[ixj];
        int   i1 = si[tid], i2 = si[ixj];
        bool up = ((tid & k) == 0);
        bool gt = (z1 > z2) || (z1 == z2 && i1 > i2);
        if (up ? gt : !gt) {
          sz[tid] = z2; sz[ixj] = z1;
          si[tid] = i2; si[ixj] = i1;
        }
      }
    }
  }
  __syncthreads();

  // --- gather into depth-sorted SoA in workspace ---
  const int g = si[i];
#pragma unroll
  for (int aidx = 0; aidx < NARR; ++aidx)
    ws[aidx*NG + i] = sp[aidx][g];
}

// ---------------------------------------------------------------------------
// Kernel 2: tiled front-to-back compositor.
// 512 threads = 16 waves; wave w owns the 16 pixels of tile row w.
// The sorted Gaussian table (26 KB) is staged into LDS once per block with
// GLOBAL_LOAD_ASYNC_TO_LDS_B128 (ASYNCcnt path, no VGPR round-trip).
// Color accumulation uses V_WMMA_F32_16X16X4_F32:
//   D(16 px x 16 ch) += A(16 px x 4 gaussians) * B(4 gaussians x 16 ch)
// A layout (f32 16x4): lane m holds A[m][0..1], lane m+16 holds A[m][2..3];
// one shfl_xor(16) carries the transmittance chain across the half-waves.
// ---------------------------------------------------------------------------
__device__ __forceinline__ float eval_alpha(const float g[NARR][NG], int k,
                                            float px, float py)
{
  const float mx = g[0][k], my = g[1][k];
  const float i00 = g[2][k], i01 = g[3][k], i11 = g[4][k];
  const float op = g[5][k];
  const bool inb = (px >= g[9][k]) & (px < g[10][k]) &
                   (py >= g[11][k]) & (py < g[12][k]);
  const float dx = px - mx, dy = py - my;
  const float mah = i00*dx*dx + 2.0f*i01*dx*dy + i11*dy*dy;
  const float al = op * __expf(-0.5f * mah);
  return inb ? al : 0.0f;
}

__global__ __launch_bounds__(512) void raster_kernel(
    const float* __restrict__ ws, float* __restrict__ out)
{
  __shared__ __align__(16) float g[NARR][NG];
  const int tid = threadIdx.x;

#if HAVE_ASYNC_LDS
  {
    // Direct global -> LDS copy, 16 bytes per lane per issue; ASYNCcnt-tracked.
    const float* src = ws;
    float* dstf = &g[0][0];
    for (int q = tid; q < (NARR * NG) / 4; q += 512) {
      __builtin_amdgcn_global_load_async_to_lds_b128(
          (__attribute__((address_space(1))) v4i_vs*)(src + q * 4),
          (__attribute__((address_space(3))) v4i_vs*)(dstf + q * 4),
          /*imm_offset=*/0, /*cpol=*/0);
    }
#if __has_builtin(__builtin_amdgcn_s_wait_asynccnt)
    __builtin_amdgcn_s_wait_asynccnt(0);
#else
    asm volatile("s_wait_asynccnt 0" ::: "memory");
#endif
  }
  __syncthreads();
#else
#pragma unroll
  for (int aidx = 0; aidx < NARR; ++aidx)
    g[aidx][tid] = ws[aidx*NG + tid];
  __syncthreads();
#endif

  const int lane = tid & 31;
  const int wave = tid >> 5;       // tile row 0..15
  const int n    = lane & 15;      // pixel-x within tile / WMMA column
  const int half = lane >> 4;      // which 2 gaussians of the chunk
  const int tile_x = blockIdx.x * 16;
  const int tile_y = blockIdx.y * 16;
  const float px = (float)(tile_x + n);
  const float py = (float)(tile_y + wave);

  v8f acc = {};
  float T = 1.0f;

  for (int cb = 0; cb < NG; cb += 4) {
    const int k0 = cb + 2*half;
    const int k1 = k0 + 1;
    const float a0 = eval_alpha(g, k0, px, py);
    const float a1 = eval_alpha(g, k1, px, py);

    if (__ballot((a0 > 0.0f) || (a1 > 0.0f)) != 0ull) {
      const float f0 = 1.0f - a0, f1 = 1.0f - a1;
      const float prod  = f0 * f1;
      const float oprod = __shfl_xor(prod, 16, 32);   // other half-wave's (1-a)(1-a)
      const float Tin = half ? T * oprod : T;          // transmittance entering k0
      v2f amat;
      amat.x = Tin * a0;            // contrib of k0 at this pixel
      amat.y = Tin * f0 * a1;       // contrib of k1
      T = T * prod * oprod;         // T after all 4 gaussians (consistent in wave)

      // B: colors of this half-wave's two gaussians; columns 3..15 are zero.
      v2f bmat;
      bmat.x = (n < 3) ? g[6 + n][k0] : 0.0f;
      bmat.y = (n < 3) ? g[6 + n][k1] : 0.0f;

#if HAVE_WMMA_F32X4
      acc = __builtin_amdgcn_wmma_f32_16x16x4_f32(
          /*neg_a=*/false, amat, /*neg_b=*/false, bmat,
          /*c_mod=*/(short)0, acc, /*reuse_a=*/false, /*reuse_b=*/false);
#else
      acc[0] += amat.x * bmat.x + amat.y * bmat.y;   // placeholder fallback
#endif
    }
    if (__ballot(T > 1e-7f) == 0ull) break;   // wave fully saturated
  }

  // D layout: VGPR r, lanes 0-15 -> (pixel M=r, channel N=lane);
  //           lanes 16-31 -> (pixel M=r+8, channel N=lane-16).
  if (n < 3) {
#pragma unroll
    for (int r = 0; r < 8; ++r) {
      const int m  = r + 8*half;                 // pixel-x within tile
      const int gx = tile_x + m;
      const int gy = tile_y + wave;
      out[(gy * IMG_W + gx) * 3 + n] = acc[r];   // BG is zero
    }
  }
}

// ---------------------------------------------------------------------------
extern "C" void kernel_launch(void* const* d_in, const int* in_sizes, int n_in,
                              void* d_out, int out_size, void* d_ws, size_t ws_size,
                              hipStream_t stream)
{
  (void)in_sizes; (void)n_in; (void)out_size; (void)ws_size;
  const float* pos   = (const float*)d_in[0];
  const float* cov3d = (const float*)d_in[1];
  const float* opac  = (const float*)d_in[2];
  const float* col   = (const float*)d_in[3];
  const float* Km    = (const float*)d_in[4];
  const float* Rm    = (const float*)d_in[5];
  const float* tv    = (const float*)d_in[6];
  float* ws  = (float*)d_ws;
  float* out = (float*)d_out;

  hipLaunchKernelGGL(preprocess_sort_kernel, dim3(1), dim3(NG), 0, stream,
                     pos, cov3d, opac, col, Km, Rm, tv, ws);
  hipLaunchKernelGGL(raster_kernel, dim3(IMG_W/16, IMG_H/16), dim3(512), 0, stream,
                     ws, out);
}